// SparseDense_89258010345880
// MI455X (gfx1250) — compile-verified
//
#include <hip/hip_runtime.h>

typedef __attribute__((ext_vector_type(2))) float v2f;
typedef __attribute__((ext_vector_type(8))) float v8f;

#define NNZ 64
#define NOUT 256
#define ROWS_PER_BLOCK 8

// One wave32 handles one sparse batch row:
//   out[row, :] = relu( sum_k vals[k] * W[idx[k], :] + Bias )
// via 16 N-tiles x 16 K-chunks of V_WMMA_F32_16X16X4_F32 (A row M=0 carries the
// sparse values; B carries four gathered W rows per chunk). 32-bit element
// offsets are precomputed once so every gather lowers to a saddr-form
// global_load_b32 with the per-tile displacement folded into the immediate.
__global__ __launch_bounds__(256) void SparseDense_wmma_kernel(
    const int* __restrict__ x_idx, const float* __restrict__ x_vals,
    const float* __restrict__ W, const float* __restrict__ Bias,
    float* __restrict__ out)
{
    const unsigned tid  = threadIdx.x;
    const unsigned lane = tid & 31u;
    const unsigned wave = tid >> 5;
    const unsigned row  = blockIdx.x * ROWS_PER_BLOCK + wave;

    const unsigned m    = lane & 15u;        // M (A) / N-within-tile (B,D) index
    const unsigned hi   = (lane >> 4) & 1u;  // lane half: K=0,1 vs K=2,3
    const bool     am   = (m == 0u);         // only A row M=0 is non-zero

    // Preload the row's 64 (value, index) pairs arranged per the 16x4 f32
    // A/B VGPR layout: lanes 0-15 take K-slots {4k, 4k+1}, lanes 16-31 take
    // {4k+2, 4k+3}. Offsets are 32-bit element indices (W has 34.8M elements).
    v2f      a[16];
    unsigned off0[16], off1[16];
    const int*   ip = x_idx  + (size_t)row * NNZ;
    const float* vp = x_vals + (size_t)row * NNZ;
#pragma unroll
    for (int k = 0; k < 16; ++k) {
        const unsigned off = 4u * (unsigned)k + 2u * hi;
        int2   rr = *(const int2*)  (ip + off);
        float2 vv = *(const float2*)(vp + off);
        a[k].x = am ? vv.x : 0.0f;
        a[k].y = am ? vv.y : 0.0f;
        off0[k] = (unsigned)rr.x * NOUT + m;
        off1[k] = (unsigned)rr.y * NOUT + m;
    }

    float*       op = out  + (size_t)row * NOUT + m;  // store base (lane<16)
    const float* bp = Bias + m;                       // bias base

#pragma unroll
    for (int n = 0; n < 16; ++n) {
        const float bias = bp[n * 16];
        v8f c = {};
#pragma unroll
        for (int k = 0; k < 16; ++k) {
            v2f b;
            b.x = W[off0[k] + n * 16];  // B[K=0|2, col] (64B/half-wave segment)
            b.y = W[off1[k] + n * 16];  // B[K=1|3, col]
            // 8 args: (neg_a, A, neg_b, B, c_mod, C, reuse_a, reuse_b)
            c = __builtin_amdgcn_wmma_f32_16x16x4_f32(
                    false, a[k], false, b, (short)0, c, false, false);
        }
        // D row M=0 lives in accumulator VGPR0, lanes 0-15 (N = lane).
        float o = c[0] + bias;
        o = o > 0.0f ? o : 0.0f;
        if (lane < 16u) op[n * 16] = o;
    }
}

extern "C" void kernel_launch(void* const* d_in, const int* in_sizes, int n_in,
                              void* d_out, int out_size, void* d_ws, size_t ws_size,
                              hipStream_t stream) {
    (void)n_in; (void)out_size; (void)d_ws; (void)ws_size;
    const int*   x_idx  = (const int*)  d_in[0];
    const float* x_vals = (const float*)d_in[1];
    const float* W      = (const float*)d_in[2];
    const float* Bias   = (const float*)d_in[3];
    float*       out    = (float*)d_out;

    const int batch = in_sizes[0] / NNZ;           // 4096
    dim3 grid((unsigned)(batch / ROWS_PER_BLOCK)); // 512 blocks
    dim3 block(256);                               // 8 wave32 / block
    hipLaunchKernelGGL(SparseDense_wmma_kernel, grid, block, 0, stream,
                       x_idx, x_vals, W, Bias, out);
}